// InvariantLayer_89893665506066
// MI455X (gfx1250) — compile-verified
//
#include <hip/hip_runtime.h>
#include <hip/hip_bf16.h>
#include <stdint.h>

// InvariantLayer: per 3x3 {F, C} compute [I1, J, -J, I4, I5], B = 4M elements.
// Memory-bound (0.43 flop/byte) -> pure 23.3 TB/s HBM play:
//   - CDNA5 async global->LDS b128 copies (ASYNCcnt), double-buffered across
//     tiles with *counted* s_wait_asynccnt waits (prefetch t+1 during compute t)
//   - conflict-free LDS staging (element strides 9 and 5 floats, coprime to 64 banks)
//   - coalesced b128 output stores via LDS-staged [TILE,5] slab.
// WMMA intentionally unused: no GEMM structure, matrix pipes cannot help here.

#define TILE  256   // elements per tile
#define BLOCK 256   // threads per block (8 waves)
#define TPB   4     // tiles per block (pipelined)

__device__ __forceinline__ void issue_async_tile(const float* __restrict__ gF,
                                                 const float* __restrict__ gC,
                                                 const float* bF, const float* bC,
                                                 int tid)
{
    const int CHUNKS = (TILE * 9 * 4) / 16;  // 576 x 16B per array
    const uint32_t ldsF = (uint32_t)(uintptr_t)bF;
    const uint32_t ldsC = (uint32_t)(uintptr_t)bC;
    for (int j = tid; j < CHUNKS; j += BLOCK) {
        const uint32_t off = (uint32_t)j * 16u;
        // GVS mode: mem = SGPR64 base + VGPR32 offset; first operand = LDS byte addr.
        asm volatile("global_load_async_to_lds_b128 %0, %1, %2"
                     :: "v"(ldsF + off), "v"(off), "s"(gF) : "memory");
        asm volatile("global_load_async_to_lds_b128 %0, %1, %2"
                     :: "v"(ldsC + off), "v"(off), "s"(gC) : "memory");
    }
    // Per-wave async instructions issued: tid<64 -> 3 loop iters -> 6, else 4
    // (wave-uniform: the 64 boundary is wave-aligned).
}

__global__ __launch_bounds__(BLOCK)
void InvariantLayer_89893665506066_kernel(const float* __restrict__ F,
                                          const float* __restrict__ C,
                                          float* __restrict__ out,
                                          int B)
{
    __shared__ __align__(16) float sF[2][TILE * 9];  // 2 x 9216 B
    __shared__ __align__(16) float sC[2][TILE * 9];  // 2 x 9216 B
    __shared__ __align__(16) float sO[TILE * 5];     // 5120 B

    const int tid    = threadIdx.x;
    const int base   = blockIdx.x * (TILE * TPB);
    const int remAll = B - base;
    int nt = (remAll + TILE - 1) / TILE;
    if (nt > TPB) nt = TPB;

    // --- prime the pipeline: tile 0 ---
    const int n0 = remAll < TILE ? remAll : TILE;
    if (n0 == TILE)
        issue_async_tile(F + (size_t)base * 9, C + (size_t)base * 9,
                         sF[0], sC[0], tid);

    for (int t = 0; t < nt; ++t) {
        const int buf  = t & 1;
        const int e0   = base + t * TILE;            // first element of this tile
        const int r    = B - e0;
        const int n    = r < TILE ? r : TILE;

        // --- prefetch tile t+1 into the other buffer (full tiles only) ---
        bool pf = false;
        if (t + 1 < nt) {
            const int r1 = B - (e0 + TILE);
            if (r1 >= TILE) {
                issue_async_tile(F + (size_t)(e0 + TILE) * 9,
                                 C + (size_t)(e0 + TILE) * 9,
                                 sF[buf ^ 1], sC[buf ^ 1], tid);
                pf = true;
            }
        }

        // --- make tile t resident in LDS ---
        if (n == TILE) {
            // Counted drain: async loads complete in order per wave, so waiting
            // down to the prefetch's in-flight count guarantees tile t is done.
            if (pf) {
                if (tid < 64) asm volatile("s_wait_asynccnt 6" ::: "memory");
                else          asm volatile("s_wait_asynccnt 4" ::: "memory");
            } else {
                asm volatile("s_wait_asynccnt 0" ::: "memory");
            }
        } else {
            // Partial tail tile (never async-prefetched): synchronous staging.
            for (int j = tid; j < n * 9; j += BLOCK) {
                sF[buf][j] = F[(size_t)e0 * 9 + j];
                sC[buf][j] = C[(size_t)e0 * 9 + j];
            }
        }
        __syncthreads();   // all waves' tile-t data visible

        // --- compute invariants ---
        if (tid < n) {
            const float* f = &sF[buf][tid * 9];   // stride 9: bank-conflict-free
            const float* c = &sC[buf][tid * 9];
            const float f00 = f[0], f01 = f[1], f02 = f[2];
            const float f10 = f[3], f11 = f[4], f12 = f[5];
            const float f20 = f[6], f21 = f[7], f22 = f[8];
            const float c00 = c[0], c01 = c[1], c02 = c[2];
            const float c10 = c[3], c11 = c[4], c12 = c[5];
            const float c20 = c[6], c21 = c[7], c22 = c[8];

            const float I1 = c00 + c11 + c22;
            const float J  = f00 * (f11 * f22 - f12 * f21)
                           - f01 * (f10 * f22 - f12 * f20)
                           + f02 * (f10 * f21 - f11 * f20);
            // G = diag(4, 0.5, 0.5):  I4 = tr(C G)
            const float I4 = 4.0f * c00 + 0.5f * (c11 + c22);
            // Cof(C) diagonal (== diag of det(C)*C^-1):  I5 = tr(Cof(C) G)
            const float M00 = c11 * c22 - c12 * c21;
            const float M11 = c00 * c22 - c02 * c20;
            const float M22 = c00 * c11 - c01 * c10;
            const float I5  = 4.0f * M00 + 0.5f * (M11 + M22);

            sO[tid * 5 + 0] = I1;                 // stride 5: conflict-free
            sO[tid * 5 + 1] = J;
            sO[tid * 5 + 2] = -J;
            sO[tid * 5 + 3] = I4;
            sO[tid * 5 + 4] = I5;
        }
        __syncthreads();   // sO ready; all reads of sF/sC[buf] done (t+2 reuse safe)

        // --- coalesced flush of the staged [TILE,5] slab ---
        if (n == TILE) {
            const int OC = (TILE * 5 * 4) / 16;   // 320 x b128
            float4* go = (float4*)(out + (size_t)e0 * 5);
            const float4* lo = (const float4*)sO;
            for (int j = tid; j < OC; j += BLOCK) go[j] = lo[j];
        } else {
            for (int j = tid; j < n * 5; j += BLOCK)
                out[(size_t)e0 * 5 + j] = sO[j];
        }
        // No barrier needed here: next compute writes sO only after the next
        // pre-compute __syncthreads(), which each wave reaches only after
        // finishing this store loop.
    }
}

extern "C" void kernel_launch(void* const* d_in, const int* in_sizes, int n_in,
                              void* d_out, int out_size, void* d_ws, size_t ws_size,
                              hipStream_t stream) {
    const float* F = (const float*)d_in[0];
    const float* C = (const float*)d_in[1];
    float* out = (float*)d_out;
    const int B = in_sizes[0] / 9;
    const int grid = (B + TILE * TPB - 1) / (TILE * TPB);
    InvariantLayer_89893665506066_kernel<<<grid, BLOCK, 0, stream>>>(F, C, out, B);
}